// Sparse_SAGEConv_26834955665988
// MI455X (gfx1250) — compile-verified
//
#include <hip/hip_runtime.h>

typedef __attribute__((ext_vector_type(2))) float v2f;
typedef __attribute__((ext_vector_type(8))) float v8f;

#define N_NODES 100000
#define N_EDGES 3200000
#define IN_CH   256
#define OUT_CH  256

// ---------------------------------------------------------------------------
// Stage 0: zero-fill (out accumulator + degree array)
// ---------------------------------------------------------------------------
__global__ __launch_bounds__(256) void zero_kernel(float* __restrict__ p, long long n) {
  long long i = (long long)blockIdx.x * blockDim.x + threadIdx.x;
  long long stride = (long long)gridDim.x * blockDim.x;
  for (; i < n; i += stride) p[i] = 0.0f;
}

// ---------------------------------------------------------------------------
// Stage 1: h = x @ W + bias   (fp32-exact via V_WMMA_F32_16X16X4_F32)
// grid.x = N_NODES/16 = 6250 blocks, 256 threads = 8 waves.
// Wave w computes the 16-row tile x two 16-col tiles: cols [16w,16w+16) and
// [128+16w, 128+16w+16), reusing the A fragment for both WMMAs.
// ---------------------------------------------------------------------------
__global__ __launch_bounds__(256) void gemm_kernel(const float* __restrict__ x,
                                                   const float* __restrict__ W,
                                                   const float* __restrict__ bias,
                                                   float* __restrict__ h) {
  const int lane = threadIdx.x & 31;
  const int wave = threadIdx.x >> 5;          // 0..7
  const int rowBase = blockIdx.x * 16;

  const int m    = lane & 15;                 // row within tile (A), col within tile (B)
  const int ko   = (lane >> 4) << 1;          // K sub-offset: 0 (lanes 0-15) or 2 (lanes 16-31)
  const int col0 = wave * 16 + m;             // first col tile
  const int col1 = 128 + wave * 16 + m;       // second col tile

  const float* __restrict__ xrow = x + (size_t)(rowBase + m) * IN_CH;

  v8f c0 = {};
  v8f c1 = {};

  #pragma unroll 8
  for (int k = 0; k < IN_CH; k += 4) {
    // A fragment (16x4): lane m holds x[row][k+ko], x[row][k+ko+1]
    v2f a = *(const v2f*)(xrow + k + ko);
    // B fragments (4x16): lane m holds W[k+ko][col], W[k+ko+1][col]
    const float* __restrict__ wp = W + (size_t)(k + ko) * OUT_CH;
    v2f b0, b1;
    b0.x = wp[col0];
    b0.y = wp[OUT_CH + col0];
    b1.x = wp[col1];
    b1.y = wp[OUT_CH + col1];
    c0 = __builtin_amdgcn_wmma_f32_16x16x4_f32(false, a, false, b0, (short)0, c0, false, false);
    c1 = __builtin_amdgcn_wmma_f32_16x16x4_f32(false, a, false, b1, (short)0, c1, false, false);
  }

  // C/D layout: VGPR v -> M = v + 8*(lane>=16), N = lane&15
  const int mhalf = (lane >> 4) << 3;         // 0 or 8
  const float bia0 = bias[col0];
  const float bia1 = bias[col1];
  #pragma unroll
  for (int v = 0; v < 8; ++v) {
    const size_t row = (size_t)(rowBase + mhalf + v);
    h[row * OUT_CH + col0] = c0[v] + bia0;
    h[row * OUT_CH + col1] = c1[v] + bia1;
  }
}

// ---------------------------------------------------------------------------
// Stage 2: edge scatter: out[dst] += h[src]; deg[dst] += 1
// One wave per edge; each lane handles 8 contiguous channels.
// h (102.4 MB) fits in the 192 MB L2, so gathers hit L2; atomics run at L2.
// ---------------------------------------------------------------------------
__global__ __launch_bounds__(256) void edge_scatter_kernel(const int* __restrict__ src,
                                                           const int* __restrict__ dst,
                                                           const float* __restrict__ h,
                                                           float* __restrict__ out,
                                                           float* __restrict__ deg) {
  const int lane = threadIdx.x & 31;
  const int wave = threadIdx.x >> 5;
  const long long e = (long long)blockIdx.x * 8 + wave;
  if (e >= N_EDGES) return;

  const int s = src[e];
  const int d = dst[e];

  const float4* __restrict__ hp = (const float4*)(h + (size_t)s * OUT_CH + lane * 8);
  float4 v0 = hp[0];
  float4 v1 = hp[1];

  float* __restrict__ op = out + (size_t)d * OUT_CH + lane * 8;
  atomicAdd(op + 0, v0.x);
  atomicAdd(op + 1, v0.y);
  atomicAdd(op + 2, v0.z);
  atomicAdd(op + 3, v0.w);
  atomicAdd(op + 4, v1.x);
  atomicAdd(op + 5, v1.y);
  atomicAdd(op + 6, v1.z);
  atomicAdd(op + 7, v1.w);

  if (lane == 0) atomicAdd(deg + d, 1.0f);
}

// ---------------------------------------------------------------------------
// Stage 3: out[i][c] /= max(deg[i], 1)
// ---------------------------------------------------------------------------
__global__ __launch_bounds__(256) void normalize_kernel(float* __restrict__ out,
                                                        const float* __restrict__ deg) {
  const size_t i = (size_t)blockIdx.x * 256 + threadIdx.x;   // exactly N_NODES*OUT_CH threads
  const float dval = fmaxf(deg[i >> 8], 1.0f);               // OUT_CH == 256
  out[i] = out[i] / dval;
}

// ---------------------------------------------------------------------------
extern "C" void kernel_launch(void* const* d_in, const int* in_sizes, int n_in,
                              void* d_out, int out_size, void* d_ws, size_t ws_size,
                              hipStream_t stream) {
  const float* x    = (const float*)d_in[0];
  const int*   src  = (const int*)d_in[1];
  const int*   dst  = (const int*)d_in[2];
  const float* W    = (const float*)d_in[3];
  const float* bias = (const float*)d_in[4];
  float* out = (float*)d_out;

  // Workspace layout: h [N_NODES*OUT_CH floats] | deg [N_NODES floats]
  float* h   = (float*)d_ws;
  float* deg = h + (size_t)N_NODES * OUT_CH;

  const long long out_elems = (long long)N_NODES * OUT_CH;   // 25,600,000

  zero_kernel<<<2048, 256, 0, stream>>>(out, out_elems);
  zero_kernel<<<512, 256, 0, stream>>>(deg, (long long)N_NODES);

  gemm_kernel<<<N_NODES / 16, 256, 0, stream>>>(x, W, bias, h);

  edge_scatter_kernel<<<N_EDGES / 8, 256, 0, stream>>>(src, dst, h, out, deg);

  normalize_kernel<<<(unsigned)(out_elems / 256), 256, 0, stream>>>(out, deg);
}